// GCN_13297218748796
// MI455X (gfx1250) — compile-verified
//
#include <hip/hip_runtime.h>
#include <hip/hip_bf16.h>

typedef __attribute__((ext_vector_type(2))) float v2f;
typedef __attribute__((ext_vector_type(8))) float v8f;

// ---------------- monotonic float <-> uint encoding for atomicMax ----------
__device__ __forceinline__ unsigned gat_encf(float x) {
    unsigned b = __float_as_uint(x);
    return (b & 0x80000000u) ? ~b : (b | 0x80000000u);
}
__device__ __forceinline__ float gat_decf(unsigned k) {
    unsigned b = (k & 0x80000000u) ? (k & 0x7FFFFFFFu) : ~k;
    return __uint_as_float(b);
}

// ---------------- WMMA f32 GEMM: C[M,Nout] = A[M,K] @ B[K,Nout] (+bias)(+relu)
// One wave per 16x16 output tile, K stepped by 4 via V_WMMA_F32_16X16X4_F32.
// M, Nout multiples of 16; K multiple of 4. EXEC all-ones per executing wave.
__global__ __launch_bounds__(128) void gat_gemm_wmma_f32(
    const float* __restrict__ A, const float* __restrict__ B,
    const float* __restrict__ bias, float* __restrict__ C,
    int M, int K, int Nout, int fuse /*0:none 1:+bias 2:+bias+relu*/)
{
    const int ntn   = Nout >> 4;
    const int total = (M >> 4) * ntn;
    const int wave  = (int)(blockIdx.x * 4 + (threadIdx.x >> 5));
    if (wave >= total) return;                 // uniform per-wave branch

    const int lane = threadIdx.x & 31;
    const int half = lane >> 4;                // 0: K+{0,1}, 1: K+{2,3}
    const int l    = lane & 15;
    const int tm   = wave / ntn, tn = wave % ntn;
    const int m0   = tm << 4,    n0 = tn << 4;

    const float* __restrict__ arow = A + (size_t)(m0 + l) * K + 2 * half;
    const float* __restrict__ bcol = B + (size_t)(2 * half) * Nout + (n0 + l);

    v8f acc = {0.f, 0.f, 0.f, 0.f, 0.f, 0.f, 0.f, 0.f};
    for (int k = 0; k < K; k += 4) {
        v2f a; a.x = arow[k];                 a.y = arow[k + 1];
        v2f b; b.x = bcol[(size_t)k * Nout];  b.y = bcol[(size_t)(k + 1) * Nout];
        acc = __builtin_amdgcn_wmma_f32_16x16x4_f32(
            false, a, false, b, (short)0, acc, false, false);
    }

    const int col = n0 + l;
    const float bv = fuse ? bias[col] : 0.f;
    float* crow = C + (size_t)(m0 + 8 * half) * Nout + col;
#pragma unroll
    for (int v = 0; v < 8; ++v) {
        float x = acc[v] + bv;
        if (fuse == 2) x = fmaxf(x, 0.f);
        crow[(size_t)v * Nout] = x;
    }
}

// ---------------- per-node attention scalars: alpha = h . a ----------------
__global__ __launch_bounds__(256) void gat_alpha(
    const float* __restrict__ hs, const float* __restrict__ hd,
    const float* __restrict__ a_s, const float* __restrict__ a_d,
    float* __restrict__ als, float* __restrict__ ald, int n, int H)
{
    int i = (int)(blockIdx.x * blockDim.x + threadIdx.x);
    if (i >= n) return;
    const float* ps = hs + (size_t)i * H;
    const float* pd = hd + (size_t)i * H;
    float ss = 0.f, sd = 0.f;
    for (int f = 0; f < H; ++f) { ss += ps[f] * a_s[f]; sd += pd[f] * a_d[f]; }
    als[i] = ss; ald[i] = sd;
}

// ---------------- generic fill (used to zero scratch each layer) -----------
__global__ __launch_bounds__(256) void gat_fill_u32(unsigned* __restrict__ p,
                                                    unsigned v, size_t n)
{
    size_t i = (size_t)blockIdx.x * blockDim.x + threadIdx.x;
    if (i < n) p[i] = v;
}

__device__ __forceinline__ float gat_leaky(float x) {
    return x > 0.f ? x : 0.2f * x;
}

// ---------------- segment max of edge logits over dst ----------------------
__global__ __launch_bounds__(256) void gat_edge_max(
    const int* __restrict__ src, const int* __restrict__ dst,
    const float* __restrict__ als, const float* __restrict__ ald,
    unsigned* __restrict__ menc, int E)
{
    int e = (int)(blockIdx.x * blockDim.x + threadIdx.x);
    if (e >= E) return;
    int s = src[e], d = dst[e];
    float lg = gat_leaky(als[s] + ald[d]);
    atomicMax(&menc[d], gat_encf(lg));
}

// ---------------- segment sum of exp(logit - max) over dst -----------------
__global__ __launch_bounds__(256) void gat_edge_expsum(
    const int* __restrict__ src, const int* __restrict__ dst,
    const float* __restrict__ als, const float* __restrict__ ald,
    const unsigned* __restrict__ menc, float* __restrict__ ssum, int E)
{
    int e = (int)(blockIdx.x * blockDim.x + threadIdx.x);
    if (e >= E) return;
    int s = src[e], d = dst[e];
    float lg = gat_leaky(als[s] + ald[d]);
    float ex = expf(lg - gat_decf(menc[d]));
    atomicAdd(&ssum[d], ex);
}

// ---------------- weighted scatter-add of source messages ------------------
// One wave32 per edge; lane f handles features 4f..4f+3 (H==128).
__global__ __launch_bounds__(256) void gat_edge_scatter(
    const int* __restrict__ src, const int* __restrict__ dst,
    const float* __restrict__ als, const float* __restrict__ ald,
    const unsigned* __restrict__ menc, const float* __restrict__ ssum,
    const float* __restrict__ hs, float* __restrict__ acc, int E, int H)
{
    int e = (int)(blockIdx.x * (blockDim.x >> 5) + (threadIdx.x >> 5));
    if (e >= E) return;
    int lane = threadIdx.x & 31;
    int s = src[e], d = dst[e];
    float lg = gat_leaky(als[s] + ald[d]);
    float w  = expf(lg - gat_decf(menc[d])) / (ssum[d] + 1e-16f);

    const float4* hp = (const float4*)(hs + (size_t)s * H);
    float*        ap = acc + (size_t)d * H;
    float4 v = hp[lane];
    int f = lane << 2;
    atomicAdd(ap + f + 0, w * v.x);
    atomicAdd(ap + f + 1, w * v.y);
    atomicAdd(ap + f + 2, w * v.z);
    atomicAdd(ap + f + 3, w * v.w);
}

// ---------------- h = relu(acc + b), in place -------------------------------
__global__ __launch_bounds__(256) void gat_bias_relu(
    float* __restrict__ h, const float* __restrict__ b, size_t total, int H)
{
    size_t i = (size_t)blockIdx.x * blockDim.x + threadIdx.x;
    if (i >= total) return;
    int f = (int)(i & (size_t)(H - 1));        // H is a power of two
    h[i] = fmaxf(h[i] + b[f], 0.f);
}

extern "C" void kernel_launch(void* const* d_in, const int* in_sizes, int n_in,
                              void* d_out, int out_size, void* d_ws, size_t ws_size,
                              hipStream_t stream)
{
    const int D = 128, H = 128, O = 64;
    const float* x   = (const float*)d_in[0];
    const int*   ei  = (const int*)d_in[1];
    const float* W1s = (const float*)d_in[2];
    const float* W1d = (const float*)d_in[3];
    const float* a1s = (const float*)d_in[4];
    const float* a1d = (const float*)d_in[5];
    const float* b1  = (const float*)d_in[6];
    const float* W2  = (const float*)d_in[7];
    const float* a2s = (const float*)d_in[8];
    const float* a2d = (const float*)d_in[9];
    const float* b2  = (const float*)d_in[10];
    const float* W3  = (const float*)d_in[11];
    const float* a3s = (const float*)d_in[12];
    const float* a3d = (const float*)d_in[13];
    const float* b3  = (const float*)d_in[14];
    const float* Wl1 = (const float*)d_in[15];
    const float* bl1 = (const float*)d_in[16];
    const float* Wl2 = (const float*)d_in[17];
    const float* bl2 = (const float*)d_in[18];

    const int N = in_sizes[0] / D;       // 100000 (multiple of 16)
    const int E = in_sizes[1] / 2;       // 1600000
    const int* src  = ei;
    const int* dstv = ei + E;

    float* ws = (float*)d_ws;
    const size_t NH = (size_t)N * H;
    float*    bufA = ws;                 // N*H
    float*    bufB = ws + NH;            // N*H
    float*    bufC = ws + 2 * NH;        // N*H
    float*    als  = ws + 3 * NH;        // N
    float*    ald  = als + N;            // N
    unsigned* menc = (unsigned*)(ald + N);   // N
    float*    ssum = (float*)(menc + N);     // N

    auto gemm = [&](const float* Ain, const float* Bin, const float* bias,
                    float* Cout, int M, int K, int Nout, int fuse) {
        int tiles  = (M >> 4) * (Nout >> 4);
        int blocks = (tiles + 3) / 4;
        gat_gemm_wmma_f32<<<blocks, 128, 0, stream>>>(Ain, Bin, bias, Cout,
                                                      M, K, Nout, fuse);
    };

    auto gat_layer = [&](const float* xin, const float* Ws, const float* Wd,
                         const float* as_, const float* ad_, const float* bb,
                         float* hs, float* hd, float* out) {
        gemm(xin, Ws, nullptr, hs, N, H, H, 0);
        if (Wd != Ws) gemm(xin, Wd, nullptr, hd, N, H, H, 0);
        gat_alpha<<<(N + 255) / 256, 256, 0, stream>>>(hs, hd, as_, ad_,
                                                       als, ald, N, H);
        gat_fill_u32<<<(N + 255) / 256, 256, 0, stream>>>(menc, 0u, (size_t)N);
        gat_fill_u32<<<(N + 255) / 256, 256, 0, stream>>>((unsigned*)ssum, 0u, (size_t)N);
        gat_fill_u32<<<(unsigned)((NH + 255) / 256), 256, 0, stream>>>((unsigned*)out, 0u, NH);
        gat_edge_max<<<(E + 255) / 256, 256, 0, stream>>>(src, dstv, als, ald, menc, E);
        gat_edge_expsum<<<(E + 255) / 256, 256, 0, stream>>>(src, dstv, als, ald,
                                                             menc, ssum, E);
        gat_edge_scatter<<<(E + 7) / 8, 256, 0, stream>>>(src, dstv, als, ald,
                                                          menc, ssum, hs, out, E, H);
        gat_bias_relu<<<(unsigned)((NH + 255) / 256), 256, 0, stream>>>(out, bb, NH, H);
    };

    // layer 1: separate src/dst linears
    gat_layer(x,    W1s, W1d, a1s, a1d, b1, bufA, bufB, bufC);   // h1 -> bufC
    // layers 2/3: shared linear (hs == hd)
    gat_layer(bufC, W2,  W2,  a2s, a2d, b2, bufA, bufA, bufB);   // h2 -> bufB
    gat_layer(bufB, W3,  W3,  a3s, a3d, b3, bufA, bufA, bufC);   // h3 -> bufC

    // MLP head
    gemm(bufC, Wl1, bl1, bufA, N, H, H, 2);                      // relu(h3@Wl1+bl1)
    gemm(bufA, Wl2, bl2, (float*)d_out, N, H, O, 1);             // @Wl2 + bl2
}